// MultinomialCrossEntropyLoss_65335042506926
// MI455X (gfx1250) — compile-verified
//
#include <hip/hip_runtime.h>

// -------------------------------------------------------------------------
// MultinomialCrossEntropyLoss for colorization (Q=313 bins, top-5 soft encode)
// MI455X plan:
//   K1 (WMMA): per-pixel d2 to all centers via V_WMMA_F32_16X16X4_F32 using
//       the embedding  d2 = [-2cx,-2cy,|c|^2,1] . [a,b,1,|ab|^2]  (K=4 exact).
//       Tile loop fully unrolled (20 WMMAs) so the compiler can overlap the
//       matrix-pipe WMMAs with the VALU top-5 insertion chain (XDL/VALU
//       co-execution). Register-resident top-5 per lane + __shfl_xor lane-pair
//       merge. Writes 5 idx + 5 w + wsum + pixel_w per pixel (~14 MB ws).
//   K2 (HBM-bound, ~369 MB stream @ 23.3 TB/s => ~16 us floor): coalesced
//       online logsumexp (1 exp/elem) fused with sparse 5-term dot; block
//       tree-reduction to partials (deterministic, no float atomics).
//   K3: single-block reduction of partials -> mean -> d_out[0].
// -------------------------------------------------------------------------

#define QBINS   313
#define HWPIX   9216          // 96*96 (H==W==96 per reference)
#define NTILES  ((QBINS + 15) / 16)   // 20
#define INV2SG2 0.02f         // 1/(2*sigma^2), sigma=5

typedef float v2f __attribute__((ext_vector_type(2)));
typedef float v8f __attribute__((ext_vector_type(8)));

__device__ __forceinline__ void top5_insert(float d, int idx, float td[5], int ti[5]) {
    if (d < td[4]) {
        td[4] = d; ti[4] = idx;
#pragma unroll
        for (int j = 4; j > 0; --j) {
            if (td[j] < td[j - 1]) {
                float fd = td[j]; td[j] = td[j - 1]; td[j - 1] = fd;
                int   fi = ti[j]; ti[j] = ti[j - 1]; ti[j - 1] = fi;
            }
        }
    }
}

// ---------------- Kernel 1: WMMA soft-encode ------------------------------
__global__ __launch_bounds__(256) void soft_encode_kernel(
    const float* __restrict__ target_ab,     // (B,2,H,W)
    const float* __restrict__ ab_centers,    // (Q,2)
    const float* __restrict__ class_weights, // (Q)
    int*   __restrict__ ws_idx,              // [5*M]
    float* __restrict__ ws_w,                // [5*M]
    float* __restrict__ ws_pw,               // [M]
    float* __restrict__ ws_wsum,             // [M]
    int M)
{
    const int lane = threadIdx.x & 31;
    const int wave = threadIdx.x >> 5;
    const int g    = blockIdx.x * 8 + wave;          // 16-pixel group per wave
    const int ngroups = (M + 15) >> 4;
    if (g >= ngroups) return;                        // wave-uniform exit

    const bool hi = lane >= 16;
    const int  lp = lane & 15;
    const int  m  = g * 16 + lp;
    const int  mc = (m < M) ? m : (M - 1);
    const int  b   = mc / HWPIX;
    const int  rem = mc - b * HWPIX;

    const float av = target_ab[(size_t)b * 2 * HWPIX + rem]         * 128.f;
    const float bv = target_ab[(size_t)b * 2 * HWPIX + HWPIX + rem] * 128.f;

    // B operand (4x16, pixels): lanes 0-15 carry rows K=0,1 = (a,b),
    // lanes 16-31 carry rows K=2,3 = (1, |ab|^2) of the same pixel (lane&15).
    v2f Bm;
    Bm.x = hi ? 1.0f : av;
    Bm.y = hi ? (av * av + bv * bv) : bv;

    float td[5]; int ti[5];
#pragma unroll
    for (int k = 0; k < 5; ++k) { td[k] = 3.0e38f; ti[k] = 0; }

    // Fully unrolled: 20 independent WMMAs; compiler can issue tile t+1's
    // center loads + v_wmma while tile t's insertion selects run on the VALU.
#pragma unroll
    for (int t = 0; t < NTILES; ++t) {
        const int  c     = t * 16 + lp;              // center row this lane feeds
        const bool valid = c < QBINS;
        const float cx = valid ? ab_centers[2 * c]     : 0.f;
        const float cy = valid ? ab_centers[2 * c + 1] : 0.f;
        const float c2 = valid ? (cx * cx + cy * cy)   : 1.0e30f; // pad -> huge d2

        // A operand (16x4, centers): lanes 0-15 rows K=0,1 = (-2cx,-2cy),
        // lanes 16-31 rows K=2,3 = (|c|^2, 1).
        v2f Am;
        Am.x = hi ? c2   : (-2.f * cx);
        Am.y = hi ? 1.0f : (-2.f * cy);

        v8f D = {0.f, 0.f, 0.f, 0.f, 0.f, 0.f, 0.f, 0.f};
        // D[center][pixel] = |ab|^2 + |c|^2 - 2*ab.c  (exact, K=4)
        D = __builtin_amdgcn_wmma_f32_16x16x4_f32(
                false, Am, false, Bm, (short)0, D, false, false);

        // D layout: lane<16 holds center rows t*16+0..7 for pixel (lane),
        //           lane>=16 holds rows t*16+8..15 for pixel (lane-16).
        const int cbase = t * 16 + (hi ? 8 : 0);
#pragma unroll
        for (int r = 0; r < 8; ++r) {
            float d2 = D[r];
            d2 = d2 < 0.f ? 0.f : d2;                // maximum(d2, 0)
            top5_insert(d2, cbase + r, td, ti);
        }
    }

    // Lane-pair merge: lane L and L+16 each hold a sorted top-5 over disjoint
    // halves of the center set for the same pixel.
    float od[5]; int oi[5];
#pragma unroll
    for (int k = 0; k < 5; ++k) {
        od[k] = __shfl_xor(td[k], 16, 32);
        oi[k] = __shfl_xor(ti[k], 16, 32);
    }
    float md[5]; int mi[5];
#pragma unroll
    for (int k = 0; k < 5; ++k) {
        const bool own = td[0] <= od[0];
        md[k] = own ? td[0] : od[0];
        mi[k] = own ? ti[0] : oi[0];
        if (own) {
#pragma unroll
            for (int j = 0; j < 4; ++j) { td[j] = td[j + 1]; ti[j] = ti[j + 1]; }
            td[4] = 3.3e38f;
        } else {
#pragma unroll
            for (int j = 0; j < 4; ++j) { od[j] = od[j + 1]; oi[j] = oi[j + 1]; }
            od[4] = 3.3e38f;
        }
    }

    // Gaussian soft weights, normalized with the reference's +1e-8.
    float wv[5]; float s = 0.f;
#pragma unroll
    for (int k = 0; k < 5; ++k) { wv[k] = __expf(-md[k] * INV2SG2); s += wv[k]; }
    const float inv   = 1.f / (s + 1e-8f);
    const float wsumN = s * inv;
    const int   nearest = mi[0];                     // argmin == top-1
    const float pw = class_weights[nearest];
#pragma unroll
    for (int k = 0; k < 5; ++k) wv[k] *= inv;

    // Sort the 5 (idx, w) pairs by index so K2 can test in ascending order.
#pragma unroll
    for (int p = 0; p < 4; ++p)
#pragma unroll
        for (int q = 0; q < 4; ++q)
            if (q < 4 - p && mi[q] > mi[q + 1]) {
                int   xi = mi[q]; mi[q] = mi[q + 1]; mi[q + 1] = xi;
                float xw = wv[q]; wv[q] = wv[q + 1]; wv[q + 1] = xw;
            }

    if (!hi && m < M) {
#pragma unroll
        for (int k = 0; k < 5; ++k) {
            ws_idx[(size_t)k * M + m] = mi[k];
            ws_w  [(size_t)k * M + m] = wv[k];
        }
        ws_pw[m]   = pw;
        ws_wsum[m] = wsumN;
    }
}

// ---------------- Kernel 2: streaming CE (HBM-bound) ----------------------
__global__ __launch_bounds__(256) void ce_reduce_kernel(
    const float* __restrict__ logits,   // (B,Q,H,W)
    const int*   __restrict__ ws_idx,
    const float* __restrict__ ws_w,
    const float* __restrict__ ws_pw,
    const float* __restrict__ ws_wsum,
    float* __restrict__ partials,
    int M)
{
    const int m = blockIdx.x * 256 + threadIdx.x;
    float val = 0.f;
    if (m < M) {
        const int b   = m / HWPIX;
        const int rem = m - b * HWPIX;
        const float* p = logits + (size_t)b * QBINS * HWPIX + rem; // stride HWPIX over q
        int ik[5]; float wk[5];
#pragma unroll
        for (int k = 0; k < 5; ++k) {
            ik[k] = ws_idx[(size_t)k * M + m];
            wk[k] = ws_w  [(size_t)k * M + m];
        }
        float mx = -3.0e38f, s = 0.f, dot = 0.f;
        for (int q = 0; q < QBINS; ++q) {
            const float x = p[(size_t)q * HWPIX];                  // coalesced
            if (q + 16 < QBINS)
                __builtin_prefetch(p + (size_t)(q + 16) * HWPIX, 0, 1);
            if (x > mx) { s = s * __expf(mx - x) + 1.f; mx = x; }  // 1 exp/elem
            else        { s += __expf(x - mx); }
#pragma unroll
            for (int k = 0; k < 5; ++k)
                dot += (q == ik[k]) ? wk[k] * x : 0.f;
        }
        const float lse = mx + __logf(s);
        const float ce  = lse * ws_wsum[m] - dot;   // -(sum w*(x - lse))
        val = ce * ws_pw[m];
    }
    __shared__ float red[256];
    red[threadIdx.x] = val;
    __syncthreads();
#pragma unroll
    for (int off = 128; off > 0; off >>= 1) {
        if ((int)threadIdx.x < off) red[threadIdx.x] += red[threadIdx.x + off];
        __syncthreads();
    }
    if (threadIdx.x == 0) partials[blockIdx.x] = red[0];
}

// ---------------- Kernel 3: deterministic final mean ----------------------
__global__ __launch_bounds__(256) void finalize_kernel(
    const float* __restrict__ partials, int n, float* __restrict__ out, int M)
{
    __shared__ float red[256];
    float s = 0.f;
    for (int i = threadIdx.x; i < n; i += 256) s += partials[i];
    red[threadIdx.x] = s;
    __syncthreads();
#pragma unroll
    for (int off = 128; off > 0; off >>= 1) {
        if ((int)threadIdx.x < off) red[threadIdx.x] += red[threadIdx.x + off];
        __syncthreads();
    }
    if (threadIdx.x == 0) out[0] = red[0] / (float)M;
}

// -------------------------------------------------------------------------
extern "C" void kernel_launch(void* const* d_in, const int* in_sizes, int n_in,
                              void* d_out, int out_size, void* d_ws, size_t ws_size,
                              hipStream_t stream) {
    const float* pred_logits   = (const float*)d_in[0];  // (B,Q,H,W)
    const float* target_ab     = (const float*)d_in[1];  // (B,2,H,W)
    const float* ab_centers    = (const float*)d_in[2];  // (Q,2)
    const float* class_weights = (const float*)d_in[3];  // (Q)
    const int M = in_sizes[1] / 2;                       // B*H*W

    // Workspace layout (floats/ints, 4B each):
    //   [0,5M)    idx   [5M,10M) w   [10M,11M) pw   [11M,12M) wsum   [12M,..) partials
    char* ws = (char*)d_ws;
    int*   ws_idx   = (int*)  (ws);
    float* ws_w     = (float*)(ws + (size_t)5  * M * 4);
    float* ws_pw    = (float*)(ws + (size_t)10 * M * 4);
    float* ws_wsum  = (float*)(ws + (size_t)11 * M * 4);
    float* partials = (float*)(ws + (size_t)12 * M * 4);

    const int ngroups = (M + 15) / 16;
    const int blocks1 = (ngroups + 7) / 8;               // 8 waves/block, 1 group/wave
    soft_encode_kernel<<<blocks1, 256, 0, stream>>>(
        target_ab, ab_centers, class_weights, ws_idx, ws_w, ws_pw, ws_wsum, M);

    const int blocks2 = (M + 255) / 256;
    ce_reduce_kernel<<<blocks2, 256, 0, stream>>>(
        pred_logits, ws_idx, ws_w, ws_pw, ws_wsum, partials, M);

    finalize_kernel<<<1, 256, 0, stream>>>(partials, blocks2, (float*)d_out, M);
}